// ClusterSelection_33981781246398
// MI455X (gfx1250) — compile-verified
//
#include <hip/hip_runtime.h>
#include <stdint.h>

// ---------------------------------------------------------------------------
// Bond-percolation flood fill, bit-packed (1 bit/site), L2-resident iteration.
// Layout: bit b of word (y*256 + w) = site (y, 32*w + b).
//   hbits[y][w] bit b : open bond (y, 32w+b) -- (y, 32w+b+1 mod 8192)
//   vbits[y][w] bit b : open bond (y, 32w+b) -- (y+1 mod 8192, 32w+b)
// Torus boundary handled by wrap in y and by h bit31 of word 255 in x.
// Tiles are staged LDS<->global with the Tensor Data Mover (TDM).
// ---------------------------------------------------------------------------

#define GRID_N 8192
#define WPR    256                  // 32-bit words per row
#define NWORDS (GRID_N * WPR)       // 2M words per bit plane (8 MB)
#define TROWS  16                   // tile rows per workgroup
#define OUTER  64                   // outer propagate launches
#define MAX_SWEEPS 16               // LDS fixed-point sweeps per launch

typedef unsigned int u32x4 __attribute__((ext_vector_type(4)));
typedef int          i32x8 __attribute__((ext_vector_type(8)));
typedef int          i32x4 __attribute__((ext_vector_type(4)));

// 4 bool-bytes (0/1) in one dword -> 4-bit nibble, carry-free multiply gather.
__device__ __forceinline__ unsigned nib4(unsigned d) {
  return (((d & 0x01010101u) * 0x01020408u) >> 24) & 0xFu;
}

// Issue one TDM 2D-tile transfer: TROWS x WPR u32 words, row stride WPR words.
// Descriptor per CDNA5 ISA 8.3/8.4 (data_size=4B). 6-arg builtin on this
// toolchain (trailing zero-filled i32x8 group). EXEC-independent; call from
// one wave only.
__device__ __forceinline__ void tdm_tile_2d(bool store, unsigned ldsOff,
                                            const unsigned* gptr) {
  uint64_t gaddr = (uint64_t)(uintptr_t)gptr;
  u32x4 g0;
  g0.x = 1u;                                            // count=1, user-mode
  g0.y = ldsOff;                                        // lds_addr (bytes)
  g0.z = (unsigned)(gaddr & 0xFFFFFFFFu);               // global_addr[31:0]
  g0.w = (unsigned)((gaddr >> 32) & 0x01FFFFFFu) | (2u << 30); // [56:32]|type=2
  i32x8 g1 = 0;
  g1[0] = (int)(2u << 16);                              // data_size=4B; mask=0
  g1[1] = (int)((WPR & 0xFFFF) << 16);                  // tensor_dim0 lo16
  g1[2] = (int)((WPR >> 16) | ((TROWS & 0xFFFF) << 16)); // dim0 hi | dim1 lo
  g1[3] = (int)((TROWS >> 16) | (WPR << 16));           // dim1 hi | tile_dim0
  g1[4] = (int)TROWS;                                   // tile_dim1|tile_dim2=0
  g1[5] = (int)WPR;                                     // dim0_stride lo32
  g1[6] = 0;
  g1[7] = 0;
  i32x4 g2 = 0, g3 = 0;                                 // 2D: groups 2/3 unused
  i32x8 g4 = 0;                                         // zero trailing group
  if (store) __builtin_amdgcn_tensor_store_from_lds(g0, g1, g2, g3, g4, 0);
  else       __builtin_amdgcn_tensor_load_to_lds(g0, g1, g2, g3, g4, 0);
}

__global__ __launch_bounds__(256) void pack_links_kernel(
    const unsigned char* __restrict__ links,
    unsigned* __restrict__ hbits, unsigned* __restrict__ vbits) {
  int t = blockIdx.x * blockDim.x + threadIdx.x;      // one 32-site word
  size_t byteoff = (size_t)t * 32u;
  const size_t plane = (size_t)GRID_N * (size_t)GRID_N;
  const uint4* vp = reinterpret_cast<const uint4*>(links + byteoff);          // links[0]
  const uint4* hp = reinterpret_cast<const uint4*>(links + plane + byteoff);  // links[1]
  uint4 v0 = vp[0], v1 = vp[1];
  uint4 h0 = hp[0], h1 = hp[1];
  unsigned vw = nib4(v0.x)        | (nib4(v0.y) << 4)  | (nib4(v0.z) << 8)  |
                (nib4(v0.w) << 12)| (nib4(v1.x) << 16) | (nib4(v1.y) << 20) |
                (nib4(v1.z) << 24)| (nib4(v1.w) << 28);
  unsigned hw = nib4(h0.x)        | (nib4(h0.y) << 4)  | (nib4(h0.z) << 8)  |
                (nib4(h0.w) << 12)| (nib4(h1.x) << 16) | (nib4(h1.y) << 20) |
                (nib4(h1.z) << 24)| (nib4(h1.w) << 28);
  vbits[t] = vw;
  hbits[t] = hw;
}

__global__ __launch_bounds__(256) void init_sel_kernel(
    unsigned* __restrict__ sel, const int* __restrict__ seed) {
  int t = blockIdx.x * blockDim.x + threadIdx.x;
  int y0 = seed[0], x0 = seed[1];
  unsigned wrd = 0u;
  if (t == y0 * WPR + (x0 >> 5)) wrd = 1u << (x0 & 31);
  sel[t] = wrd;
}

__global__ __launch_bounds__(256) void propagate_kernel(
    const unsigned* __restrict__ hbits,
    const unsigned* __restrict__ vbits,
    unsigned* __restrict__ sel) {
  __shared__ unsigned sh_sel[TROWS + 2][WPR];   // rows ty-1 .. ty+TROWS (halo)
  __shared__ unsigned sh_h[TROWS][WPR];         // rows ty   .. ty+TROWS-1
  __shared__ unsigned sh_v[TROWS + 1][WPR];     // rows ty-1 .. ty+TROWS-1
  __shared__ int sh_changed;

  const int w  = threadIdx.x;                   // word column 0..255 (full row)
  const int ty = blockIdx.x * TROWS;

  // --- TDM: stage the three rectangular tiles (h, v, sel body rows) async.
  if (threadIdx.x == 0) {
    tdm_tile_2d(false, (unsigned)(uintptr_t)(&sh_h[0][0]),
                hbits + (size_t)ty * WPR);
    tdm_tile_2d(false, (unsigned)(uintptr_t)(&sh_v[1][0]),
                vbits + (size_t)ty * WPR);
    tdm_tile_2d(false, (unsigned)(uintptr_t)(&sh_sel[1][0]),
                sel + (size_t)ty * WPR);
    __builtin_amdgcn_s_wait_tensorcnt(0);
  }

  // Torus-wrapped halo rows via plain loads (not rectangular in general).
  {
    int yup = (ty - 1) & (GRID_N - 1);
    int ydn = (ty + TROWS) & (GRID_N - 1);
    sh_sel[0][w]         = sel[(size_t)yup * WPR + w];
    sh_sel[TROWS + 1][w] = sel[(size_t)ydn * WPR + w];
    sh_v[0][w]           = vbits[(size_t)yup * WPR + w];
  }
  __syncthreads();

  // Iterate tile to local fixed point. Monotone (bits only set), so the
  // benign LDS races across waves converge to the same closure.
  for (int it = 0; it < MAX_SWEEPS; ++it) {
    if (threadIdx.x == 0) sh_changed = 0;
    __syncthreads();
    unsigned any = 0;
    const int up = it & 1;                      // alternate sweep direction
    for (int k = 0; k < TROWS; ++k) {
      const int r = up ? (TROWS - k) : (1 + k);
      unsigned s = sh_sel[r][w];
      const unsigned old = s;
      const unsigned hh  = sh_h[r - 1][w];
      const int wl = (w + WPR - 1) & (WPR - 1);
      const int wr = (w + 1) & (WPR - 1);
      // vertical injections through open v-bonds
      s |= sh_sel[r - 1][w] & sh_v[r - 1][w];
      s |= sh_sel[r + 1][w] & sh_v[r][w];
      // cross-word injections (x torus wrap is word 255 <-> word 0)
      s |= (sh_sel[r][wl] >> 31) & (sh_h[r - 1][wl] >> 31);          // -> bit0
      s |= ((sh_sel[r][wr] & 1u) & (hh >> 31)) << 31;                // -> bit31
      // intra-word closure over contiguous open-bond runs (Kogge-Stone)
      const unsigned m1  = hh & 0x7FFFFFFFu;    // bit31 is the cross-word bond
      const unsigned m2  = m1 & (m1 >> 1);
      const unsigned m4  = m2 & (m2 >> 2);
      const unsigned m8  = m4 & (m4 >> 4);
      const unsigned m16 = m8 & (m8 >> 8);
      s |= (s & m1)  << 1;  s |= (s & m2)  << 2;  s |= (s & m4) << 4;
      s |= (s & m8)  << 8;  s |= (s & m16) << 16;
      s |= (s >> 1)  & m1;  s |= (s >> 2)  & m2;  s |= (s >> 4) & m4;
      s |= (s >> 8)  & m8;  s |= (s >> 16) & m16;
      sh_sel[r][w] = s;
      any |= (s ^ old);
    }
    if (any) atomicOr(&sh_changed, 1);
    __syncthreads();
    if (sh_changed == 0) break;                 // uniform across the block
    __syncthreads();                            // protect flag reset vs read
  }

  // Write back owned rows via TDM store (halo rows are read-only). All LDS
  // writes are visible after the barrier; s_endpgm's implicit wait-idle plus
  // the explicit tensorcnt wait keep the transfer inside kernel lifetime.
  __syncthreads();
  if (threadIdx.x == 0) {
    tdm_tile_2d(true, (unsigned)(uintptr_t)(&sh_sel[1][0]),
                sel + (size_t)ty * WPR);
    __builtin_amdgcn_s_wait_tensorcnt(0);
  }
}

__global__ __launch_bounds__(256) void expand_kernel(
    const unsigned* __restrict__ sel, float4* __restrict__ out) {
  int t = blockIdx.x * blockDim.x + threadIdx.x;   // one 32-site word
  unsigned wrd = sel[t];
  float4* p = out + (size_t)t * 8;                 // 32 floats = 8 x float4
  #pragma unroll
  for (int q = 0; q < 8; ++q) {
    p[q] = make_float4((float)((wrd >> (4 * q + 0)) & 1u),
                       (float)((wrd >> (4 * q + 1)) & 1u),
                       (float)((wrd >> (4 * q + 2)) & 1u),
                       (float)((wrd >> (4 * q + 3)) & 1u));
  }
}

extern "C" void kernel_launch(void* const* d_in, const int* in_sizes, int n_in,
                              void* d_out, int out_size, void* d_ws, size_t ws_size,
                              hipStream_t stream) {
  (void)in_sizes; (void)n_in; (void)out_size; (void)ws_size;
  const unsigned char* links = (const unsigned char*)d_in[0]; // bool[2,8192,8192]
  const int* seed            = (const int*)d_in[1];           // int32[2]
  float4* out                = (float4*)d_out;                // float[8192,8192]

  // Workspace: 3 bit planes, 8 MB each (24 MB total, L2-resident).
  unsigned* hb  = (unsigned*)d_ws;
  unsigned* vb  = hb + NWORDS;
  unsigned* sl  = vb + NWORDS;

  pack_links_kernel<<<NWORDS / 256, 256, 0, stream>>>(links, hb, vb);
  init_sel_kernel<<<NWORDS / 256, 256, 0, stream>>>(sl, seed);
  for (int i = 0; i < OUTER; ++i) {
    propagate_kernel<<<GRID_N / TROWS, 256, 0, stream>>>(hb, vb, sl);
  }
  expand_kernel<<<NWORDS / 256, 256, 0, stream>>>(sl, out);
}